// GATConv_51084341018875
// MI455X (gfx1250) — compile-verified
//
#include <hip/hip_runtime.h>

// GAT forward for MI455X (gfx1250, wave32).
// Roofline: ~0.93 GB total HBM traffic (dominated by 839 MB x_lifted writes)
// -> ~40us at 23.3 TB/s. GEMM (3.3 GFLOP) is noise; done with fp32 WMMA
// (V_WMMA_F32_16X16X4_F32) to match fp32 reference precision. h (51.2 MB)
// lives in workspace and stays resident in the 192 MB L2 so edge gathers hit
// L2; all big output streams use non-temporal stores to avoid evicting h.

typedef float v2f __attribute__((ext_vector_type(2)));
typedef float v4f __attribute__((ext_vector_type(4)));
typedef float v8f __attribute__((ext_vector_type(8)));

#define N_NODES 100000
#define N_EDGES 1638400
#define IN_CH   128
#define HEADS   4
#define OUT_CH  32
#define NOUT    (HEADS * OUT_CH)   // 128

// ---------------------------------------------------------------------------
// Kernel 1: h = x @ W^T + b  (M=100000, N=128, K=128), fp32 WMMA 16x16x4.
// Block = 256 threads = 8 waves. Each wave owns one 16-col N-tile of a
// 16-row M-strip. K-loop: 32 steps of one v_wmma_f32_16x16x4_f32.
//
// A (16x4 f32) lane layout (ISA 7.12.2): lanes 0-15 -> M=lane, VGPR0=K0,
// VGPR1=K1; lanes 16-31 -> M=lane-16, VGPR0=K2, VGPR1=K3. B (4x16) is the
// mirrored layout with lanes indexing N. Both become contiguous float2 loads.
// C/D (16x16 f32): VGPR r -> M = r + 8*(lane/16), N = lane%16.
// ---------------------------------------------------------------------------
__global__ __launch_bounds__(256)
void gat_gemm_wmma(const float* __restrict__ x,
                   const float* __restrict__ W,
                   const float* __restrict__ bias,
                   float* __restrict__ h) {
  const int lane = threadIdx.x & 31;
  const int wave = threadIdx.x >> 5;        // 0..7 -> N tile
  const int row0 = blockIdx.x * 16;         // M tile origin
  const int n0   = wave * 16;
  const int half = lane >> 4;               // 0: K0/K1, 1: K2/K3
  const int l16  = lane & 15;

  const float* xp = x + (size_t)(row0 + l16) * IN_CH + 2 * half;
  const float* wp = W + (size_t)(n0   + l16) * IN_CH + 2 * half;

  v8f c = {};
#pragma unroll
  for (int k = 0; k < IN_CH; k += 4) {
    v2f a = *(const v2f*)(xp + k);   // x[m][k + 2*half .. +1]
    v2f b = *(const v2f*)(wp + k);   // W[n][k + 2*half .. +1]
    // 8 args: (neg_a, A, neg_b, B, c_mod, C, reuse_a, reuse_b)
    c = __builtin_amdgcn_wmma_f32_16x16x4_f32(
        false, a, false, b, (short)0, c, false, false);
  }

  const int   n  = n0 + l16;
  const float bv = bias[n];
  float* out = h + (size_t)(row0 + 8 * half) * NOUT + n;
#pragma unroll
  for (int r = 0; r < 8; ++r) {
    out[(size_t)r * NOUT] = c[r] + bv;
  }
}

// ---------------------------------------------------------------------------
// Kernel 2: per-node attention scores. One wave per node. Lane l loads
// h[node][4l..4l+3] (float4, fully coalesced 512B row) and the matching
// att_l/att_r float4 (L1-resident). Head = l/8; 8-lane shfl_xor reduction.
// ---------------------------------------------------------------------------
__global__ __launch_bounds__(256)
void gat_alpha(const float* __restrict__ h,
               const float* __restrict__ att_l,
               const float* __restrict__ att_r,
               float* __restrict__ al,
               float* __restrict__ ar) {
  const int lane = threadIdx.x & 31;
  const int wave = threadIdx.x >> 5;
  const int node = blockIdx.x * 8 + wave;
  if (node >= N_NODES) return;

  const v4f hv  = *(const v4f*)(h + (size_t)node * NOUT + lane * 4);
  const v4f atl = *(const v4f*)(att_l + lane * 4);
  const v4f atr = *(const v4f*)(att_r + lane * 4);

  float sl = hv.x * atl.x + hv.y * atl.y + hv.z * atl.z + hv.w * atl.w;
  float sr = hv.x * atr.x + hv.y * atr.y + hv.z * atr.z + hv.w * atr.w;

#pragma unroll
  for (int off = 1; off < 8; off <<= 1) {
    sl += __shfl_xor(sl, off, 32);
    sr += __shfl_xor(sr, off, 32);
  }
  if ((lane & 7) == 0) {
    const int head = lane >> 3;
    al[node * HEADS + head] = sl;
    ar[node * HEADS + head] = sr;
  }
}

// ---------------------------------------------------------------------------
// Kernel 3: per-edge gather + leaky_relu. One wave per edge: 32 lanes copy
// the 128-float h row as float4s (512B contiguous store per edge). h reads
// use default temporal hints (L2-resident, 51 MB < 192 MB L2); all output
// stores are non-temporal to keep the 865 MB write stream out of L2.
// ---------------------------------------------------------------------------
__global__ __launch_bounds__(256)
void gat_edges(const int* __restrict__ src_idx,
               const int* __restrict__ trg_idx,
               const float* __restrict__ h,
               const float* __restrict__ al,
               const float* __restrict__ ar,
               float* __restrict__ out_alpha,
               float* __restrict__ out_x) {
  const int lane = threadIdx.x & 31;
  const int wave = threadIdx.x >> 5;
  const size_t e = (size_t)blockIdx.x * 8 + wave;
  if (e >= (size_t)N_EDGES) return;

  const int s = src_idx[e];
  const int t = trg_idx[e];

  const v4f v = *(const v4f*)(h + (size_t)s * NOUT + lane * 4);
  __builtin_nontemporal_store(v, (v4f*)(out_x + e * NOUT) + lane);

  if (lane < HEADS) {
    float a = al[s * HEADS + lane] + ar[t * HEADS + lane];
    a = (a > 0.0f) ? a : 0.01f * a;   // leaky_relu, slope 0.01
    __builtin_nontemporal_store(a, out_alpha + e * HEADS + lane);
  }
}

// ---------------------------------------------------------------------------
extern "C" void kernel_launch(void* const* d_in, const int* in_sizes, int n_in,
                              void* d_out, int out_size, void* d_ws, size_t ws_size,
                              hipStream_t stream) {
  const float* x     = (const float*)d_in[0];   // [N, 128]
  const float* W     = (const float*)d_in[1];   // [128, 128]
  const float* b     = (const float*)d_in[2];   // [128]
  const float* att_l = (const float*)d_in[3];   // [1,4,32]
  const float* att_r = (const float*)d_in[4];   // [1,4,32]
  const int*   src   = (const int*)d_in[5];     // [E]
  const int*   trg   = (const int*)d_in[6];     // [E]

  // Workspace: h [N,128] (51.2 MB) + alpha_l/alpha_r [N,4] each (1.6 MB each)
  float* h  = (float*)d_ws;
  float* al = h  + (size_t)N_NODES * NOUT;
  float* ar = al + (size_t)N_NODES * HEADS;

  // Outputs concatenated flat: alpha_per_edge [E,4] then x_lifted [E,4,32]
  float* out_alpha = (float*)d_out;
  float* out_x     = out_alpha + (size_t)N_EDGES * HEADS;

  gat_gemm_wmma<<<N_NODES / 16, 256, 0, stream>>>(x, W, b, h);
  gat_alpha<<<(N_NODES + 7) / 8, 256, 0, stream>>>(h, att_l, att_r, al, ar);
  gat_edges<<<N_EDGES / 8, 256, 0, stream>>>(src, trg, h, al, ar, out_alpha, out_x);
}